// HardConstrainedMLP_unroll_72937134620678
// MI455X (gfx1250) — compile-verified
//
#include <hip/hip_runtime.h>
#include <hip/hip_bf16.h>

// ---------------------------------------------------------------------------
// HardConstrainedMLP on MI455X (gfx1250, wave32).
// fp32 WMMA (V_WMMA_F32_16X16X4_F32) everywhere. The 100-iteration
// Douglas-Rachford loop is fused into one kernel with all operands
// LDS-resident (~247KB of 320KB/WGP), bank-conflict-free padded layouts,
// transposed B staging so every fragment is a single ds_load_b64, and the
// v-pass folded into the GEMM2 epilogue (2 barriers/iter).
// ---------------------------------------------------------------------------

#define SIGMA 1.0f
#define OMEGA 1.7f

typedef float v2f __attribute__((ext_vector_type(2)));
typedef float v8f __attribute__((ext_vector_type(8)));

static __device__ __forceinline__ v8f wmma_f32(v2f a, v2f b, v8f c) {
    // D = A(16x4 f32) * B(4x16 f32) + C(16x16 f32)
    return __builtin_amdgcn_wmma_f32_16x16x4_f32(
        /*neg_a=*/false, a, /*neg_b=*/false, b,
        /*c_mod=*/(short)0, c, /*reuse_a=*/false, /*reuse_b=*/false);
}

// ---------------------------------------------------------------------------
// prep: K = inv(A A^T) (Gauss-Jordan in LDS), MopT[256,64] = (K @ A)^T.
// One workgroup; negligible vs the 33 GFLOP pipeline.
// ---------------------------------------------------------------------------
__global__ void prep_kernel(const float* __restrict__ A, float* __restrict__ MopT) {
    __shared__ float Aug[64 * 130];
    const int tid = threadIdx.x;  // 256 threads

    for (int e = tid; e < 4096; e += 256) {
        const int i = e >> 6, j = e & 63;
        float s = 0.0f;
        for (int d = 0; d < 256; ++d) s = fmaf(A[i * 256 + d], A[j * 256 + d], s);
        Aug[i * 130 + j] = s;
        Aug[i * 130 + 64 + j] = (i == j) ? 1.0f : 0.0f;
    }
    __syncthreads();

    const int r = tid >> 2;
    const int cblk = (tid & 3) * 32;
    for (int p = 0; p < 64; ++p) {
        const float piv = Aug[p * 130 + p];
        const float factor = Aug[r * 130 + p];
        __syncthreads();
        if (tid < 128) Aug[p * 130 + tid] *= (1.0f / piv);
        __syncthreads();
        if (r != p) {
            for (int c = cblk; c < cblk + 32; ++c)
                Aug[r * 130 + c] -= factor * Aug[p * 130 + c];
        }
        __syncthreads();
    }

    // MopT[d,m] = sum_j K[m,j] * A[j,d]
    for (int o = tid; o < 16384; o += 256) {
        const int m = o >> 8, d = o & 255;
        float s = 0.0f;
        for (int j = 0; j < 64; ++j)
            s = fmaf(Aug[m * 130 + 64 + j], A[j * 256 + d], s);
        MopT[d * 64 + m] = s;
    }
}

// ---------------------------------------------------------------------------
// Trunk GEMM: Y[M,N] = act(X[M,K] @ W[K,N] + bias), fp32 WMMA.
// Block 256 thr (8 waves), tile 64x128, 32-deep K chunks. W staged transposed
// (sWt[n][k], stride 34) so B-frags are single ds_load_b64; sX stride 36.
// Strides 36 and 34 are padded off multiples of 32 banks -> conflict-free fragment loads.
// ---------------------------------------------------------------------------
__global__ void gemm_bias_act(const float* __restrict__ X, const float* __restrict__ W,
                              const float* __restrict__ bias, float* __restrict__ Y,
                              int M, int N, int K, int doRelu) {
    __shared__ float sX[64 * 36];    // 9 KB (padded)
    __shared__ float sWt[128 * 34];  // 17 KB (transposed+padded)

    const int tid  = threadIdx.x;
    const int lane = tid & 31;
    const int w    = tid >> 5;
    const int rb   = (w & 3) * 16;
    const int cg   = (w >> 2) * 64;
    const int lr   = lane & 15;
    const int lh   = lane >> 4;
    const int blockRow = blockIdx.x * 64;
    const int blockCol = blockIdx.y * 128;

    v8f acc[4];
    #pragma unroll
    for (int t = 0; t < 4; ++t) acc[t] = {};

    for (int k0 = 0; k0 < K; k0 += 32) {
        for (int i = tid; i < 512; i += 256) {
            const int rr = (i << 2) >> 5, cc = (i << 2) & 31;
            *(float4*)&sX[rr * 36 + cc] =
                *(const float4*)&X[(size_t)(blockRow + rr) * K + k0 + cc];
        }
        for (int i = tid; i < 1024; i += 256) {
            const int kk = (i << 2) >> 7, nn = (i << 2) & 127;
            const float4 wv = *(const float4*)&W[(size_t)(k0 + kk) * N + blockCol + nn];
            sWt[(nn + 0) * 34 + kk] = wv.x;
            sWt[(nn + 1) * 34 + kk] = wv.y;
            sWt[(nn + 2) * 34 + kk] = wv.z;
            sWt[(nn + 3) * 34 + kk] = wv.w;
        }
        if (k0 + 32 < K)
            __builtin_prefetch(&W[(size_t)(k0 + 32 + (tid >> 3)) * N + blockCol + (tid & 7) * 16], 0, 1);
        __syncthreads();

        #pragma unroll
        for (int kk = 0; kk < 32; kk += 4) {
            const int ks = kk + 2 * lh;
            const v2f af = *(const v2f*)&sX[(rb + lr) * 36 + ks];  // ds_load_b64
            #pragma unroll
            for (int t = 0; t < 4; ++t) {
                const v2f bf = *(const v2f*)&sWt[(cg + t * 16 + lr) * 34 + ks];  // ds_load_b64
                acc[t] = wmma_f32(af, bf, acc[t]);
            }
        }
        __syncthreads();
    }

    #pragma unroll
    for (int t = 0; t < 4; ++t) {
        #pragma unroll
        for (int r = 0; r < 8; ++r) {
            const int m   = r + 8 * lh;
            const int row = blockRow + rb + m;
            const int col = blockCol + cg + t * 16 + lr;
            float v = acc[t][r] + bias[col];
            if (doRelu) v = fmaxf(v, 0.0f);
            Y[(size_t)row * N + col] = v;
        }
    }
}

// ---------------------------------------------------------------------------
// Fused Douglas-Rachford loop. Block = 256 thr (8 waves) owns 32 rows.
// LDS floats: sA 64x260=16640, sMt 256x68=17408, sZ/sY0/sV 32x260=8320 ea,
// sT 32x68=2176, sB 32x64=2048  -> 63232 floats = 247 KB.
// Per iter/wave: GEMM1 (64 wmma, K=256) -> t' ; GEMM2 (64 wmma, K=64,
// A-frags register-cached) ; fused y/w/z/v epilogue. 2 barriers/iter.
// v0 == y0 exactly since z0 = y0.
// ---------------------------------------------------------------------------
__global__ void iterate_kernel(const float* __restrict__ y0, const float* __restrict__ bcon,
                               const float* __restrict__ A, const float* __restrict__ MopT,
                               const int* __restrict__ n_iter_p, float* __restrict__ out) {
    extern __shared__ float sm[];
    float* sA  = sm;              // [64][260]
    float* sMt = sA + 16640;      // [256][68]  (Mop^T)
    float* sZ  = sMt + 17408;     // [32][260]
    float* sY0 = sZ + 8320;       // [32][260]
    float* sV  = sY0 + 8320;      // [32][260]
    float* sT  = sV + 8320;       // [32][68]
    float* sB  = sT + 2176;       // [32][64]

    const int tid  = threadIdx.x;     // 0..255
    const int lane = tid & 31;
    const int w    = tid >> 5;        // 0..7
    const int band = (w >> 2) * 16;   // row band 0 / 16
    const int lr   = lane & 15;
    const int lh   = lane >> 4;
    const int rowBase = blockIdx.x * 32;

    // stage constants + 32-row state tile
    for (int i = tid; i < 4096; i += 256) {                 // A [64][256]
        const int r = (i << 2) >> 8, c = (i << 2) & 255;
        *(float4*)&sA[r * 260 + c] = *(const float4*)&A[i << 2];
    }
    for (int i = tid; i < 4096; i += 256) {                 // MopT [256][64]
        const int n = (i << 2) >> 6, k = (i << 2) & 63;
        *(float4*)&sMt[n * 68 + k] = *(const float4*)&MopT[i << 2];
    }
    for (int i = tid; i < 2048; i += 256) {                 // y0 tile [32][256]
        const int r = (i << 2) >> 8, c = (i << 2) & 255;
        const float4 v = *(const float4*)&y0[(size_t)(rowBase + r) * 256 + c];
        *(float4*)&sY0[r * 260 + c] = v;
        *(float4*)&sZ[r * 260 + c]  = v;   // z0 = y0
        *(float4*)&sV[r * 260 + c]  = v;   // v0 = (z0 + s*y0)/(1+s) = y0
    }
    for (int i = tid; i < 512; i += 256)
        *(float4*)&sB[i << 2] = *(const float4*)&bcon[(size_t)rowBase * 64 + (i << 2)];
    __syncthreads();

    const int n_iter = *n_iter_p;
    const float inv = 1.0f / (1.0f + SIGMA);
    const int tcol   = (w & 3) * 16;   // GEMM1 output column tile
    const int colgrp = (w & 3) * 64;   // GEMM2 output column group

    for (int it = 0; it < n_iter; ++it) {
        // GEMM1: t[32,64] = v @ A^T ; this wave: rows [band,band+16), cols [tcol,tcol+16)
        v8f acc = {};
        #pragma unroll 8
        for (int k0 = 0; k0 < 256; k0 += 4) {
            const int ks = k0 + 2 * lh;
            const v2f af = *(const v2f*)&sV[(band + lr) * 260 + ks];
            const v2f bf = *(const v2f*)&sA[(tcol + lr) * 260 + ks];  // A^T col = A row
            acc = wmma_f32(af, bf, acc);
        }
        #pragma unroll
        for (int r = 0; r < 8; ++r) {   // t' = t - bcon
            const int m = band + r + 8 * lh;
            const int n = tcol + lr;
            sT[m * 68 + n] = acc[r] - sB[m * 64 + n];
        }
        __syncthreads();

        // GEMM2: u[32,256] = t' @ Mop ; this wave: rows [band,+16), cols [colgrp,+64)
        v2f afk[16];
        #pragma unroll
        for (int j = 0; j < 16; ++j)
            afk[j] = *(const v2f*)&sT[(band + lr) * 68 + 4 * j + 2 * lh];
        const int last = (it == n_iter - 1);
        #pragma unroll
        for (int nt = 0; nt < 4; ++nt) {
            const int n0 = colgrp + nt * 16;
            v8f acc2 = {};
            #pragma unroll
            for (int j = 0; j < 16; ++j) {
                const v2f bf = *(const v2f*)&sMt[(n0 + lr) * 68 + 4 * j + 2 * lh];
                acc2 = wmma_f32(afk[j], bf, acc2);
            }
            // y = v - u ; w = max(2y - z, 0) ; z += OMEGA*(w - y) ; v = (z + s*y0)/(1+s)
            #pragma unroll
            for (int r = 0; r < 8; ++r) {
                const int m   = band + r + 8 * lh;
                const int nn  = n0 + lr;
                const int idx = m * 260 + nn;
                const float vv = sV[idx];
                const float zz = sZ[idx];
                const float y  = vv - acc2[r];
                const float ww = fmaxf(2.0f * y - zz, 0.0f);
                const float zn = zz + OMEGA * (ww - y);
                sZ[idx] = zn;
                sV[idx] = (zn + SIGMA * sY0[idx]) * inv;
                if (last) out[(size_t)(rowBase + m) * 256 + nn] = y;
            }
        }
        __syncthreads();
    }
}

// ---------------------------------------------------------------------------
extern "C" void kernel_launch(void* const* d_in, const int* in_sizes, int n_in,
                              void* d_out, int out_size, void* d_ws, size_t ws_size,
                              hipStream_t stream) {
    const float* x    = (const float*)d_in[0];
    const float* bcon = (const float*)d_in[1];
    const float* A    = (const float*)d_in[2];
    const float* W1   = (const float*)d_in[3];
    const float* b1   = (const float*)d_in[4];
    const float* W2   = (const float*)d_in[5];
    const float* b2   = (const float*)d_in[6];
    const float* W3   = (const float*)d_in[7];
    const float* b3   = (const float*)d_in[8];
    const float* Wout = (const float*)d_in[9];
    const float* bout = (const float*)d_in[10];
    const int*   nit  = (const int*)d_in[11];
    float* out = (float*)d_out;

    char* ws = (char*)d_ws;
    float* MopT = (float*)ws;                                 // 64 KB
    float* h1   = (float*)(ws + (64 << 10));                  // 8 MB
    float* h2   = (float*)(ws + (64 << 10) + (8u << 20));     // 8 MB
    float* y0   = (float*)(ws + (64 << 10) + (16u << 20));    // 4 MB

    prep_kernel<<<1, 256, 0, stream>>>(A, MopT);

    const dim3 blk(256);
    gemm_bias_act<<<dim3(4096 / 64, 512 / 128), blk, 0, stream>>>(x,  W1,   b1,   h1, 4096, 512, 128, 1);
    gemm_bias_act<<<dim3(4096 / 64, 512 / 128), blk, 0, stream>>>(h1, W2,   b2,   h2, 4096, 512, 512, 1);
    gemm_bias_act<<<dim3(4096 / 64, 512 / 128), blk, 0, stream>>>(h2, W3,   b3,   h1, 4096, 512, 512, 1);
    gemm_bias_act<<<dim3(4096 / 64, 256 / 128), blk, 0, stream>>>(h1, Wout, bout, y0, 4096, 256, 512, 0);

    const size_t smem = 63232 * sizeof(float);  // 247 KB < 320 KB/WGP
    iterate_kernel<<<dim3(4096 / 32), dim3(256), smem, stream>>>(y0, bcon, A, MopT, nit, out);
}